// RecSysGNN_76295799046815
// MI455X (gfx1250) — compile-verified
//
#include <hip/hip_runtime.h>
#include <stdint.h>

// ---------------------------------------------------------------------------
// RecSysGNN (DySimGCF / LightGCN-style) for MI455X (gfx1250).
// Memory-bound scatter/gather workload: WMMA adds nothing (0.92 GFLOP total vs
// ~GBs of L2 traffic). We target the CDNA5 data-movement paths instead:
//   * global_load_async_to_lds_b32 + s_wait_asynccnt (double-buffered edge
//     stream staging into LDS)
//   * global_atomic_add_f32 scope:SCOPE_DEV (device-coherent no-return HW
//     float atomics for the scatter-sum; RMW executes at the shared 192MB L2)
//   * wave32 layout: one edge per wave, float2 per lane (64 dims / 32 lanes)
// Working set (2x 38.4MB feature arrays + edge streams) is L2-resident.
// ---------------------------------------------------------------------------

#define NTHREADS 256
#define TILE     1024
#define PER_THR  (TILE / NTHREADS)          // 4 elements per thread per array
// async instructions per thread per tile = 3 arrays * PER_THR = 12
static_assert(TILE % NTHREADS == 0, "tile must divide evenly");

// flat LDS pointer's low 32 bits == LDS byte offset (shared aperture layout)
__device__ __forceinline__ uint32_t lds_off_u32(const void* p) {
  return (uint32_t)(uintptr_t)p;
}

__device__ __forceinline__ void async_gl2lds_b32(uint32_t lds_off, const void* g) {
  asm volatile("global_load_async_to_lds_b32 %0, %1, off"
               :: "v"(lds_off), "v"((unsigned long long)(uintptr_t)g)
               : "memory");
}

// Device-scope float atomic add (no return -> STOREcnt tracked).
// scope:SCOPE_DEV is required: default scope is WGP, which is not guaranteed
// coherent across WGPs for RMW ops.
__device__ __forceinline__ void gatomic_fadd(float* p, float v) {
  asm volatile("global_atomic_add_f32 %0, %1, off scope:SCOPE_DEV"
               :: "v"((unsigned long long)(uintptr_t)p), "v"(v)
               : "memory");
}

// order-preserving encode/decode of float for unsigned atomicMax
__device__ __forceinline__ uint32_t enc_ord(float f) {
  uint32_t u = __float_as_uint(f);
  return (u & 0x80000000u) ? ~u : (u | 0x80000000u);
}
__device__ __forceinline__ float dec_ord(uint32_t k) {
  uint32_t u = (k & 0x80000000u) ? (k & 0x7fffffffu) : ~k;
  return __uint_as_float(u);
}

// ---------------------------------------------------------------------------
__global__ void zero_u32_kernel(uint32_t* __restrict__ p, int n) {
  int i = blockIdx.x * blockDim.x + threadIdx.x;
  int stride = gridDim.x * blockDim.x;
  for (; i < n; i += stride) p[i] = 0u;
}

__global__ void edge_max_kernel(const float* __restrict__ attrs,
                                const int* __restrict__ frm,
                                const int* __restrict__ to,
                                uint32_t* __restrict__ m_in_key,
                                uint32_t* __restrict__ m_out_key, int E) {
  int i = blockIdx.x * blockDim.x + threadIdx.x;
  int stride = gridDim.x * blockDim.x;
  for (; i < E; i += stride) {
    uint32_t k = enc_ord(attrs[i]);
    atomicMax(&m_in_key[to[i]], k);
    atomicMax(&m_out_key[frm[i]], k);
  }
}

__global__ void edge_expsum_kernel(const float* __restrict__ attrs,
                                   const int* __restrict__ frm,
                                   const int* __restrict__ to,
                                   const uint32_t* __restrict__ m_in_key,
                                   const uint32_t* __restrict__ m_out_key,
                                   float* __restrict__ s_in,
                                   float* __restrict__ s_out, int E) {
  int i = blockIdx.x * blockDim.x + threadIdx.x;
  int stride = gridDim.x * blockDim.x;
  for (; i < E; i += stride) {
    float a = attrs[i];
    int te = to[i], fe = frm[i];
    gatomic_fadd(&s_in[te],  expf(a - dec_ord(m_in_key[te])));
    gatomic_fadd(&s_out[fe], expf(a - dec_ord(m_out_key[fe])));
  }
}

__global__ void edge_norm_kernel(const float* __restrict__ attrs,
                                 const int* __restrict__ frm,
                                 const int* __restrict__ to,
                                 const uint32_t* __restrict__ m_in_key,
                                 const uint32_t* __restrict__ m_out_key,
                                 const float* __restrict__ s_in,
                                 const float* __restrict__ s_out,
                                 float* __restrict__ nrm, int E) {
  int i = blockIdx.x * blockDim.x + threadIdx.x;
  int stride = gridDim.x * blockDim.x;
  for (; i < E; i += stride) {
    float a = attrs[i];
    int te = to[i], fe = frm[i];
    float inc  = expf(a - dec_ord(m_in_key[te]))  / s_in[te];
    float outg = expf(a - dec_ord(m_out_key[fe])) / s_out[fe];
    nrm[i] = sqrtf(inc * outg);
  }
}

__global__ void scale_init_kernel(const float* __restrict__ src,
                                  float* __restrict__ dst, int n) {
  int i = blockIdx.x * blockDim.x + threadIdx.x;
  int stride = gridDim.x * blockDim.x;
  for (; i < n; i += stride) dst[i] = 0.25f * src[i];
}

// out += 0.25*xn ; optionally zero the ping-pong buffer for the next layer
__global__ void accum_kernel(const float* __restrict__ xn,
                             float* __restrict__ out,
                             float* __restrict__ other, int n, int zero_other) {
  int i = blockIdx.x * blockDim.x + threadIdx.x;
  int stride = gridDim.x * blockDim.x;
  for (; i < n; i += stride) {
    out[i] += 0.25f * xn[i];
    if (zero_other) other[i] = 0.0f;
  }
}

// ---------------------------------------------------------------------------
// Propagation: x_next[to[e], :] += nrm[e] * x_prev[frm[e], :]
// Edge streams staged into LDS with async copies, double-buffered.
// One edge per wave; lane owns dims [2*lane, 2*lane+1].
// ---------------------------------------------------------------------------
__global__ void __launch_bounds__(NTHREADS)
scatter_kernel(const float* __restrict__ x_prev,
               const int* __restrict__ frm,
               const int* __restrict__ to,
               const float* __restrict__ nrm,
               float* __restrict__ x_next,
               int n_edges, int n_tiles) {
  __shared__ int   sh_frm[2][TILE];
  __shared__ int   sh_to [2][TILE];
  __shared__ float sh_nrm[2][TILE];

  const int tid  = threadIdx.x;
  const int wave = tid >> 5;
  const int lane = tid & 31;
  const int nwaves = NTHREADS >> 5;

  if ((int)blockIdx.x >= n_tiles) return;

  // Always issues exactly 12 async b32 copies per thread (clamped addresses),
  // so s_wait_asynccnt 12 deterministically drains the previous tile while the
  // freshly prefetched one stays in flight (async loads complete in order).
  auto issue_tile = [&](int t, int b) {
    int tt = t % n_tiles;                 // overrun prefetch stays in-bounds
    int base = tt * TILE;
#pragma unroll
    for (int k = 0; k < PER_THR; ++k) {
      int e = tid + k * NTHREADS;
      int g = base + e;
      if (g > n_edges - 1) g = n_edges - 1;   // clamp (tail slots unused)
      async_gl2lds_b32(lds_off_u32(&sh_frm[b][e]), frm + g);
      async_gl2lds_b32(lds_off_u32(&sh_to [b][e]), to  + g);
      async_gl2lds_b32(lds_off_u32(&sh_nrm[b][e]), nrm + g);
    }
  };

  int t = blockIdx.x;
  int buf = 0;
  issue_tile(t, 0);

  for (; t < n_tiles; t += gridDim.x) {
    issue_tile(t + gridDim.x, buf ^ 1);                 // prefetch next tile
    asm volatile("s_wait_asynccnt 12" ::: "memory");    // current tile landed
    __syncthreads();

    int base = t * TILE;
    int rem = n_edges - base;
    if (rem > TILE) rem = TILE;

    for (int i = wave; i < rem; i += nwaves) {
      int   f = sh_frm[buf][i];           // LDS broadcast reads
      int   d = sh_to [buf][i];
      float w = sh_nrm[buf][i];
      const float2 v = *(const float2*)(x_prev + (size_t)f * 64 + lane * 2);
      float* dst = x_next + (size_t)d * 64 + lane * 2;
      gatomic_fadd(dst,     v.x * w);
      gatomic_fadd(dst + 1, v.y * w);
    }
    __syncthreads();        // everyone done with `buf` before it is re-filled
    buf ^= 1;
  }
}

// ---------------------------------------------------------------------------
static inline int nblocks(long n) {
  long b = (n + NTHREADS - 1) / NTHREADS;
  if (b > 4096) b = 4096;
  if (b < 1) b = 1;
  return (int)b;
}

extern "C" void kernel_launch(void* const* d_in, const int* in_sizes, int n_in,
                              void* d_out, int out_size, void* d_ws, size_t ws_size,
                              hipStream_t stream) {
  (void)n_in; (void)out_size; (void)ws_size;
  const float* emb   = (const float*)d_in[0];   // [N, 64] f32
  const int*   eidx  = (const int*)  d_in[1];   // [2, E]  int32
  const float* attrs = (const float*)d_in[2];   // [E]     f32

  const int NND = in_sizes[0];        // N*64 = 9,600,000
  const int E   = in_sizes[2];        // 2,400,000
  const int N   = NND / 64;
  const int* frm = eidx;
  const int* to  = eidx + E;
  float* out = (float*)d_out;

  // workspace layout (256B aligned): bufA | bufB | nrm | stats[4N]  (~89MB)
  uintptr_t p = (uintptr_t)d_ws;
  auto take = [&](size_t bytes) {
    uintptr_t r = (p + 255) & ~(uintptr_t)255;
    p = r + bytes;
    return r;
  };
  float*    bufA  = (float*)   take((size_t)NND * 4);
  float*    bufB  = (float*)   take((size_t)NND * 4);
  float*    nrm   = (float*)   take((size_t)E * 4);
  uint32_t* stats = (uint32_t*)take((size_t)4 * N * 4);
  uint32_t* m_in_key  = stats;
  uint32_t* m_out_key = stats + N;
  float*    s_in      = (float*)(stats + 2 * (size_t)N);
  float*    s_out     = (float*)(stats + 3 * (size_t)N);

  // 1) zero softmax stats and the first accumulator
  zero_u32_kernel<<<nblocks(4L * N), NTHREADS, 0, stream>>>(stats, 4 * N);
  zero_u32_kernel<<<nblocks(NND), NTHREADS, 0, stream>>>((uint32_t*)bufA, NND);

  // 2) segment softmax stats -> per-edge norm
  edge_max_kernel   <<<nblocks(E), NTHREADS, 0, stream>>>(attrs, frm, to, m_in_key, m_out_key, E);
  edge_expsum_kernel<<<nblocks(E), NTHREADS, 0, stream>>>(attrs, frm, to, m_in_key, m_out_key, s_in, s_out, E);
  edge_norm_kernel  <<<nblocks(E), NTHREADS, 0, stream>>>(attrs, frm, to, m_in_key, m_out_key, s_in, s_out, nrm, E);

  // 3) out = 0.25 * x0
  scale_init_kernel<<<nblocks(NND), NTHREADS, 0, stream>>>(emb, out, NND);

  // 4) three propagation layers, ping-pong accumulators
  const int n_tiles = (E + TILE - 1) / TILE;
  const int sgrid = n_tiles > 2048 ? 2048 : n_tiles;

  const float* xp = emb;
  float* xn = bufA;     // zeroed above
  float* other = bufB;  // zeroed by accum of layer 0
  for (int l = 0; l < 3; ++l) {
    scatter_kernel<<<sgrid, NTHREADS, 0, stream>>>(xp, frm, to, nrm, xn, E, n_tiles);
    accum_kernel<<<nblocks(NND), NTHREADS, 0, stream>>>(xn, out, other, NND, l < 2 ? 1 : 0);
    xp = xn;
    float* t2 = xn; xn = other; other = t2;
  }
}